// Conv2d_winograd_1735166787616
// MI455X (gfx1250) — compile-verified
//
#include <hip/hip_runtime.h>
#include <hip/hip_bf16.h>

typedef __attribute__((ext_vector_type(2))) float v2f;
typedef __attribute__((ext_vector_type(8))) float v8f;

// Problem constants (from reference): B=64, Cin=Cout=256, H=W=32,
// F(4x4,3x3) Winograd: 6x6 transform domain (36 points), nT=8, tiles=64*8*8=4096.
#define NB   64
#define NC   256
#define NO   256
#define NH   32
#define NT   8
#define NTIL 4096   // NB*NT*NT
#define NP   36

// ---------------------------------------------------------------------------
// Kernel 1: weight transform  U[p][cout][cin] = (G w G^T)[p]
// ---------------------------------------------------------------------------
__global__ void wg_weight_transform(const float* __restrict__ w,
                                    float* __restrict__ U) {
  int tid  = blockIdx.x * blockDim.x + threadIdx.x;   // 65536 threads
  int cout = tid >> 8;
  int cin  = tid & 255;
  const float* wp = w + (size_t)(cout * NC + cin) * 9;
  float g[9];
#pragma unroll
  for (int i = 0; i < 9; ++i) g[i] = wp[i];

  const float G[6][3] = {
    { 0.25f,        0.f,          0.f        },
    {-1.f/6.f,     -1.f/6.f,     -1.f/6.f    },
    {-1.f/6.f,      1.f/6.f,     -1.f/6.f    },
    { 1.f/24.f,     1.f/12.f,     1.f/6.f    },
    { 1.f/24.f,    -1.f/12.f,     1.f/6.f    },
    { 0.f,          0.f,          1.f        }
  };

  float u[6][3];
#pragma unroll
  for (int r = 0; r < 6; ++r)
#pragma unroll
    for (int c = 0; c < 3; ++c)
      u[r][c] = G[r][0] * g[0 * 3 + c] + G[r][1] * g[1 * 3 + c] + G[r][2] * g[2 * 3 + c];

#pragma unroll
  for (int r = 0; r < 6; ++r)
#pragma unroll
    for (int c = 0; c < 6; ++c) {
      float val = u[r][0] * G[c][0] + u[r][1] * G[c][1] + u[r][2] * G[c][2];
      int p = r * 6 + c;
      U[((size_t)p * NO + cout) * NC + cin] = val;
    }
}

// ---------------------------------------------------------------------------
// Kernel 2: input transform  V[p][tile][cin] = (B_t d B_t^T)[p]
// grid: (ty=8, b=64); block 256 = 8 tx * 32 cin-chunk; loop 8 cin-chunks
// ---------------------------------------------------------------------------
__global__ void wg_input_transform(const float* __restrict__ x,
                                   float* __restrict__ V) {
  int ty = blockIdx.x;
  int b  = blockIdx.y;
  int tx = threadIdx.x & 7;
  int cq = threadIdx.x >> 3;             // [0,32)
  int r0 = ty * 4 - 1;
  int c0 = tx * 4 - 1;
  int tile = (b * NT + ty) * NT + tx;

  for (int cc = 0; cc < 8; ++cc) {
    int c = cc * 32 + cq;
    const float* xp = x + ((size_t)(b * NC + c) * NH) * NH;

    float d[6][6];
#pragma unroll
    for (int i = 0; i < 6; ++i) {
      int r = r0 + i;
#pragma unroll
      for (int j = 0; j < 6; ++j) {
        int col = c0 + j;
        d[i][j] = (r >= 0 && r < NH && col >= 0 && col < NH) ? xp[r * NH + col] : 0.f;
      }
    }

    // t = B_t * d  (row combinations)
    float t[6][6];
#pragma unroll
    for (int j = 0; j < 6; ++j) {
      t[0][j] =  4.f * d[0][j] - 5.f * d[2][j] + d[4][j];
      t[1][j] = -4.f * d[1][j] - 4.f * d[2][j] + d[3][j] + d[4][j];
      t[2][j] =  4.f * d[1][j] - 4.f * d[2][j] - d[3][j] + d[4][j];
      t[3][j] = -2.f * d[1][j] -       d[2][j] + 2.f * d[3][j] + d[4][j];
      t[4][j] =  2.f * d[1][j] -       d[2][j] - 2.f * d[3][j] + d[4][j];
      t[5][j] =  4.f * d[1][j] - 5.f * d[3][j] + d[5][j];
    }
    // v = t * B_t^T  (same combinations over columns)
#pragma unroll
    for (int i = 0; i < 6; ++i) {
      float v0 =  4.f * t[i][0] - 5.f * t[i][2] + t[i][4];
      float v1 = -4.f * t[i][1] - 4.f * t[i][2] + t[i][3] + t[i][4];
      float v2 =  4.f * t[i][1] - 4.f * t[i][2] - t[i][3] + t[i][4];
      float v3 = -2.f * t[i][1] -       t[i][2] + 2.f * t[i][3] + t[i][4];
      float v4 =  2.f * t[i][1] -       t[i][2] - 2.f * t[i][3] + t[i][4];
      float v5 =  4.f * t[i][1] - 5.f * t[i][3] + t[i][5];
      size_t base = ((size_t)(i * 6) * NTIL + tile) * NC + c;
      size_t str  = (size_t)NTIL * NC;
      V[base + 0 * str] = v0;
      V[base + 1 * str] = v1;
      V[base + 2 * str] = v2;
      V[base + 3 * str] = v3;
      V[base + 4 * str] = v4;
      V[base + 5 * str] = v5;
    }
  }
}

// ---------------------------------------------------------------------------
// Kernel 3: 36 batched GEMMs  M[p] = V[p] (4096x256) x U[p] (256 couts)
// grid (32 mblk, 2 nblk, 36 p), block 128 = 4 waves (2x2 over m,n).
// Each wave: 64x64 tile = 4x4 register blocking of V_WMMA_F32_16X16X4_F32.
// Per K-step: 8 b64 operand loads (2 KB) feed 16 WMMA (32768 FLOP)
//   -> 16 FLOP/byte from L2 (vs 8 with 2x2 blocking).
// A layout: lane(0-15)=M row, half-wave selects K pair {0,1}/{2,3} -> b64 load.
// B layout: lane(0-15)=N col, same K-pair split, U stored [p][n][k] -> b64.
// ---------------------------------------------------------------------------
#define WMMA4(acc, a, b) \
  acc = __builtin_amdgcn_wmma_f32_16x16x4_f32(false, a, false, b, (short)0, acc, false, false)

__global__ void __launch_bounds__(128)
wg_gemm(const float* __restrict__ Vm, const float* __restrict__ Um,
        float* __restrict__ Mm) {
  int p    = blockIdx.z;
  int lane = threadIdx.x & 31;
  int wave = threadIdx.x >> 5;                  // 0..3
  int mBase = blockIdx.x * 128 + (wave & 1) * 64;
  int nBase = blockIdx.y * 128 + (wave >> 1) * 64;

  const float* Vp = Vm + (size_t)p * NTIL * NC;
  const float* Up = Um + (size_t)p * NO * NC;

  int row = lane & 15;                          // M row / N col within 16-block
  int kh  = (lane >> 4) * 2;                    // K sub-pair select

  const float* aP = Vp + (size_t)(mBase + row) * NC + kh;
  const float* bP = Up + (size_t)(nBase + row) * NC + kh;

  v8f c00 = {}, c01 = {}, c02 = {}, c03 = {};
  v8f c10 = {}, c11 = {}, c12 = {}, c13 = {};
  v8f c20 = {}, c21 = {}, c22 = {}, c23 = {};
  v8f c30 = {}, c31 = {}, c32 = {}, c33 = {};

#pragma unroll 2
  for (int k = 0; k < NC; k += 4) {
    v2f a0 = *(const v2f*)(aP + k);
    v2f a1 = *(const v2f*)(aP + k + 16 * NC);
    v2f a2 = *(const v2f*)(aP + k + 32 * NC);
    v2f a3 = *(const v2f*)(aP + k + 48 * NC);
    v2f b0 = *(const v2f*)(bP + k);
    v2f b1 = *(const v2f*)(bP + k + 16 * NC);
    v2f b2 = *(const v2f*)(bP + k + 32 * NC);
    v2f b3 = *(const v2f*)(bP + k + 48 * NC);
    WMMA4(c00, a0, b0); WMMA4(c01, a0, b1); WMMA4(c02, a0, b2); WMMA4(c03, a0, b3);
    WMMA4(c10, a1, b0); WMMA4(c11, a1, b1); WMMA4(c12, a1, b2); WMMA4(c13, a1, b3);
    WMMA4(c20, a2, b0); WMMA4(c21, a2, b1); WMMA4(c22, a2, b2); WMMA4(c23, a2, b3);
    WMMA4(c30, a3, b0); WMMA4(c31, a3, b1); WMMA4(c32, a3, b2); WMMA4(c33, a3, b3);
  }

  // C/D layout: VGPR r, lane L -> m = r + (L>=16)*8, n = L&15
  float* Mp = Mm + (size_t)p * NTIL * NO;
  int mOff = (lane >> 4) * 8;
  int nOff = lane & 15;
  float* outBase = Mp + (size_t)(mBase + mOff) * NO + nBase + nOff;

#define STORE_ACC(acc, im, in)                                        \
  _Pragma("unroll")                                                   \
  for (int r = 0; r < 8; ++r)                                         \
    outBase[(size_t)((im) * 16 + r) * NO + (in) * 16] = (acc)[r];

  STORE_ACC(c00, 0, 0) STORE_ACC(c01, 0, 1) STORE_ACC(c02, 0, 2) STORE_ACC(c03, 0, 3)
  STORE_ACC(c10, 1, 0) STORE_ACC(c11, 1, 1) STORE_ACC(c12, 1, 2) STORE_ACC(c13, 1, 3)
  STORE_ACC(c20, 2, 0) STORE_ACC(c21, 2, 1) STORE_ACC(c22, 2, 2) STORE_ACC(c23, 2, 3)
  STORE_ACC(c30, 3, 0) STORE_ACC(c31, 3, 1) STORE_ACC(c32, 3, 2) STORE_ACC(c33, 3, 3)
#undef STORE_ACC
}

// ---------------------------------------------------------------------------
// Kernel 4: inverse transform  y = A_t pw A_t^T, scatter to NCHW output
// grid: (ty=8, b=64); block 256 = 8 tx * 32 cout-chunk; loop 8 chunks
// ---------------------------------------------------------------------------
__global__ void wg_output_transform(const float* __restrict__ Mm,
                                    float* __restrict__ out) {
  int ty = blockIdx.x;
  int b  = blockIdx.y;
  int tx = threadIdx.x & 7;
  int nq = threadIdx.x >> 3;             // [0,32)
  int tile = (b * NT + ty) * NT + tx;

  for (int nc = 0; nc < 8; ++nc) {
    int n = nc * 32 + nq;
    float pw[6][6];
#pragma unroll
    for (int p = 0; p < NP; ++p)
      pw[p / 6][p % 6] = Mm[((size_t)p * NTIL + tile) * NO + n];

    // u = A_t (4x6) * pw
    float u[4][6];
#pragma unroll
    for (int j = 0; j < 6; ++j) {
      u[0][j] = pw[0][j] + pw[1][j] + pw[2][j] + pw[3][j] + pw[4][j];
      u[1][j] = pw[1][j] - pw[2][j] + 2.f * pw[3][j] - 2.f * pw[4][j];
      u[2][j] = pw[1][j] + pw[2][j] + 4.f * pw[3][j] + 4.f * pw[4][j];
      u[3][j] = pw[1][j] - pw[2][j] + 8.f * pw[3][j] - 8.f * pw[4][j] + pw[5][j];
    }
    float* op = out + ((size_t)(b * NO + n) * NH + ty * 4) * NH + tx * 4;
#pragma unroll
    for (int i = 0; i < 4; ++i) {
      float y0 = u[i][0] + u[i][1] + u[i][2] + u[i][3] + u[i][4];
      float y1 = u[i][1] - u[i][2] + 2.f * u[i][3] - 2.f * u[i][4];
      float y2 = u[i][1] + u[i][2] + 4.f * u[i][3] + 4.f * u[i][4];
      float y3 = u[i][1] - u[i][2] + 8.f * u[i][3] - 8.f * u[i][4] + u[i][5];
      op[i * NH + 0] = y0;
      op[i * NH + 1] = y1;
      op[i * NH + 2] = y2;
      op[i * NH + 3] = y3;
    }
  }
}

// ---------------------------------------------------------------------------
// Launch. Workspace layout (floats):
//   U: 36*256*256        =  2,359,296   (9.4 MB)
//   V: 36*4096*256       = 37,748,736   (151 MB)
//   M: 36*4096*256       = 37,748,736   (151 MB)
// total 77,856,768 floats = 311.4 MB of d_ws.
// ---------------------------------------------------------------------------
extern "C" void kernel_launch(void* const* d_in, const int* in_sizes, int n_in,
                              void* d_out, int out_size, void* d_ws, size_t ws_size,
                              hipStream_t stream) {
  const float* x = (const float*)d_in[0];   // (64,256,32,32)
  const float* w = (const float*)d_in[1];   // (256,256,3,3)
  // d_in[2..4] = A_t, B_t, G -- transforms are hardcoded above.

  float* ws = (float*)d_ws;
  float* U  = ws;
  float* V  = ws + (size_t)NP * NO * NC;                          // +2,359,296
  float* M  = V  + (size_t)NP * NTIL * NC;                        // +37,748,736
  float* out = (float*)d_out;

  wg_weight_transform<<<dim3(NO * NC / 256), 256, 0, stream>>>(w, U);
  wg_input_transform<<<dim3(NT, NB), 256, 0, stream>>>(x, V);
  wg_gemm<<<dim3(NTIL / 128, NO / 128, NP), 128, 0, stream>>>(V, U, M);
  wg_output_transform<<<dim3(NT, NB), 256, 0, stream>>>(M, out);
}